// DTGNN_2_61478161875046
// MI455X (gfx1250) — compile-verified
//
#include <hip/hip_runtime.h>
#include <math.h>

// -------- CDNA5 (gfx1250) fused DTGNN forward --------
// One workgroup, 8 wave32s. All GEMMs via V_WMMA_F32_16X16X4_F32 (exact fp32).

typedef __attribute__((ext_vector_type(2))) float v2f;
typedef __attribute__((ext_vector_type(8))) float v8f;

#if defined(__gfx1250__) && __has_builtin(__builtin_amdgcn_wmma_f32_16x16x4_f32)
#define USE_WMMA_F32 1
#else
#define USE_WMMA_F32 0
#endif

// C[M16*16 x Ntiles*16] = A[M16*16 x K] * B[K x N], A rows zero-padded.
// Tiles distributed across the 8 waves; wave-uniform control flow (EXEC all-1s).
__device__ __forceinline__ void mm_tiles(
    const float* __restrict__ A, int lda,
    const float* __restrict__ B, int ldb,
    float* __restrict__ C, int ldc,
    int M16, int Ntiles, int K, int wave, int lane)
{
  const int total = M16 * Ntiles;
  const int colA  = lane & 15;
  const int g     = lane >> 4;
  for (int t = wave; t < total; t += 8) {
    const int mt = t / Ntiles, nt = t % Ntiles;
#if USE_WMMA_F32
    v8f acc = {};
    const float* Ar = A + (mt * 16 + colA) * lda;  // lane&15 = M row for A
    const float* Bc = B + nt * 16 + colA;          // lane&15 = N col for B
    for (int k = 0; k < K; k += 4) {
      const int ka = k + 2 * g;                    // lane group selects K pair
      v2f a, b;
      a.x = Ar[ka];           a.y = Ar[ka + 1];
      b.x = Bc[ka * ldb];     b.y = Bc[(ka + 1) * ldb];
      acc = __builtin_amdgcn_wmma_f32_16x16x4_f32(
          false, a, false, b, (short)0, acc, false, false);
    }
    float* Cr = C + (mt * 16 + 8 * g) * ldc + nt * 16 + colA;
#pragma unroll
    for (int j = 0; j < 8; ++j) Cr[j * ldc] = acc[j];
#else
    for (int j = 0; j < 8; ++j) {
      const int m = mt * 16 + 8 * g + j;
      const int n = nt * 16 + colA;
      float s = 0.f;
      for (int k = 0; k < K; ++k) s += A[m * lda + k] * B[k * ldb + n];
      C[m * ldc + n] = s;
    }
#endif
  }
}

__global__ __launch_bounds__(256) void dtgnn_fused(
    const float* __restrict__ x_feat,  const float* __restrict__ xft,
    const float* __restrict__ edge_attr,
    const float* __restrict__ c1w1, const float* __restrict__ c1b1,
    const float* __restrict__ c1w2, const float* __restrict__ c1b2,
    const float* __restrict__ g1_lin, const float* __restrict__ g1_as,
    const float* __restrict__ g1_ad,  const float* __restrict__ g1_le,
    const float* __restrict__ g1_ae,  const float* __restrict__ g1_b,
    const float* __restrict__ g2_lin, const float* __restrict__ g2_as,
    const float* __restrict__ g2_ad,  const float* __restrict__ g2_le,
    const float* __restrict__ g2_ae,  const float* __restrict__ g2_b,
    const float* __restrict__ mlp_w1, const float* __restrict__ mlp_b1,
    const float* __restrict__ mlp_w2, const float* __restrict__ mlp_b2,
    const float* __restrict__ d1w, const float* __restrict__ d1b,
    const float* __restrict__ d2w, const float* __restrict__ d2b,
    const float* __restrict__ d3w, const float* __restrict__ d3b,
    const float* __restrict__ c2w1, const float* __restrict__ c2b1,
    const float* __restrict__ c2w2, const float* __restrict__ c2b2,
    const float* __restrict__ c2l1w, const float* __restrict__ c2l1b,
    const float* __restrict__ c2l2w, const float* __restrict__ c2l2b,
    const int* __restrict__ edge_index,
    float* __restrict__ out)
{
  __shared__ float arena[12800];              // 50 KB, stage-aliased
  __shared__ float s_t2[192], s_t3[200], s_xcnn[16];
  __shared__ float s_mean[128];
  __shared__ float s_v[512];
  __shared__ int   s_src[24], s_dst[24];
  __shared__ float s_as[32], s_ad[32], s_ae[96], s_w[96], s_mx[32], s_den[32];
  __shared__ float s_vec3[192];

  const int tid  = threadIdx.x;
  const int lane = tid & 31;
  const int wave = tid >> 5;

  // arena aliasing plan (float offsets):
  float* sX1  = arena;           // [16][512]  0..8191      (stage A)
  float* sH1  = arena + 8192;    // [16][256]  8192..12287
  float* sEA1 = arena;           // [24][128]  0..3071      (after sX1 dead)
  float* sX2  = arena + 4096;    // [16][256]  4096..8191
  float* sHid = arena;           // [16][64]   0..1023
  float* sEAn = arena + 2048;    // [16][64]   2048..3071
  float* sH2  = arena;           // [16][64]   0..1023
  float* sEA2 = arena + 8192;    // [24][64]   8192..9727
  float* sX3  = arena + 9728;    // [8][64]    9728..10239
  float* sC2  = arena;           // [64][4][10] 0..2559
  float* sP   = arena + 4608;    // [64][128]  4608..12799 (pool, then flat in-place)
  float* sTmp = arena;           // [16][256]  0..4095

  // ===== Stage 1: CNN_1 =====
  if (tid < 192) {  // t2[b][c][l] padded (+1 col border), via .view(2,8,10) index math
    int b = tid / 96, r = tid % 96, c = r / 12, lc = r % 12, l = lc - 1;
    float v = 0.f;
    if (l >= 0 && l < 10)
      v = xft[((c & 1) ? 5 : 1) * 80 + (4 * b + (c >> 1)) * 10 + l];
    s_t2[tid] = v;
  }
  if (tid < 200) s_t3[tid] = 0.f;
  __syncthreads();
  if (tid < 160) {  // conv1d(8->8, k=3, pad=1) + relu
    int b = tid / 80, r = tid % 80, co = r / 10, l = r % 10;
    float s = c1b1[co];
    for (int ci = 0; ci < 8; ++ci)
      for (int k = 0; k < 3; ++k)
        s += s_t2[(b * 8 + ci) * 12 + l + k] * c1w1[co * 24 + ci * 3 + k];
    s_t3[(b * 10 + co + 1) * 10 + l] = fmaxf(s, 0.f);
  }
  __syncthreads();
  if (tid < 16) {   // conv1d over time axis (10->1, k=3, pad=1), then relu(view)
    int b = tid >> 3, n = tid & 7;
    float s = c1b2[0];
    for (int t10 = 0; t10 < 10; ++t10)
      for (int k = 0; k < 3; ++k)
        s += s_t3[(b * 10 + n + k) * 10 + t10] * c1w2[t10 * 3 + k];
    s_xcnn[tid] = fmaxf(s, 0.f);
  }
  __syncthreads();

  // ===== Stage 2: X1 = concat(x_feat, x_cnn) padded to 16 rows =====
  for (int i = tid; i < 8192; i += 256) {
    int rr = i >> 9, cc = i & 511;
    float v = 0.f;
    if (rr < 8)
      v = (cc < 510) ? x_feat[rr * 510 + cc] : s_xcnn[rr * 2 + (cc - 510)];
    sX1[i] = v;
  }
  __syncthreads();

  // ===== Stage 3: H1 = X1 @ g1_lin  [16][256]  (WMMA) =====
  mm_tiles(sX1, 512, g1_lin, 256, sH1, 256, 1, 16, 512, wave, lane);
  __syncthreads();

  // ===== Stage 4: GAT1 attention =====
  if (tid < 128) {           // column means of edge_attr (self-loop fill)
    float s = 0.f;
    for (int e = 0; e < 16; ++e) s += edge_attr[e * 128 + tid];
    s_mean[tid] = s * (1.f / 16.f);
  }
  for (int i = tid; i < 512; i += 256) {  // v1[k][hd] = (g1_le @ a_e) folded
    int k = i >> 2, hd = i & 3;
    float s = 0.f;
    for (int c = 0; c < 64; ++c)
      s += g1_le[k * 256 + hd * 64 + c] * g1_ae[hd * 64 + c];
    s_v[i] = s;
  }
  if (tid < 24) {            // src/dst with appended self loops
    s_src[tid] = (tid < 16) ? edge_index[tid]      : (tid - 16);
    s_dst[tid] = (tid < 16) ? edge_index[16 + tid] : (tid - 16);
  }
  if (tid < 32) {            // per-node a_s·h, a_d·h
    int n = tid >> 2, hd = tid & 3;
    float sa = 0.f, sd = 0.f;
    for (int c = 0; c < 64; ++c) {
      float h = sH1[n * 256 + hd * 64 + c];
      sa += h * g1_as[hd * 64 + c];
      sd += h * g1_ad[hd * 64 + c];
    }
    s_as[tid] = sa; s_ad[tid] = sd;
  }
  __syncthreads();
  for (int i = tid; i < 3072; i += 256) {  // ea_full [24][128] (sX1 dead)
    int e = i >> 7, c = i & 127;
    sEA1[i] = (e < 16) ? edge_attr[i] : s_mean[c];
  }
  __syncthreads();
  if (tid < 96) {            // alpha = leaky_relu(as[src]+ad[dst]+ea·v1)
    int e = tid >> 2, hd = tid & 3;
    float s = 0.f;
    for (int k = 0; k < 128; ++k) s += sEA1[e * 128 + k] * s_v[k * 4 + hd];
    float a = s_as[s_src[e] * 4 + hd] + s_ad[s_dst[e] * 4 + hd] + s;
    s_ae[tid] = (a > 0.f) ? a : 0.2f * a;
  }
  __syncthreads();
  if (tid < 32) {            // segment max over dst
    int n = tid >> 2, hd = tid & 3;
    float m = -3.0e38f;
    for (int e = 0; e < 24; ++e)
      if (s_dst[e] == n) m = fmaxf(m, s_ae[e * 4 + hd]);
    s_mx[tid] = m;
  }
  __syncthreads();
  if (tid < 96) {
    int e = tid >> 2, hd = tid & 3;
    s_w[tid] = __expf(s_ae[tid] - s_mx[s_dst[e] * 4 + hd]);
  }
  __syncthreads();
  if (tid < 32) {            // segment sum
    int n = tid >> 2, hd = tid & 3;
    float s = 0.f;
    for (int e = 0; e < 24; ++e)
      if (s_dst[e] == n) s += s_w[e * 4 + hd];
    s_den[tid] = s;
  }
  __syncthreads();
  if (tid < 96) {
    int e = tid >> 2, hd = tid & 3;
    s_w[tid] = s_w[tid] / (s_den[s_dst[e] * 4 + hd] + 1e-16f);
  }
  __syncthreads();
  for (int i = tid; i < 4096; i += 256) {  // aggregate -> X2 (+bias, relu), pad rows
    int n = i >> 8, c = i & 255;
    float v = 0.f;
    if (n < 8) {
      int hd = c >> 6;
      float s = 0.f;
      for (int e = 0; e < 24; ++e)
        if (s_dst[e] == n) s += s_w[e * 4 + hd] * sH1[s_src[e] * 256 + c];
      v = fmaxf(s + g1_b[c], 0.f);
    }
    sX2[i] = v;
  }
  __syncthreads();

  // ===== Stage 5: edge MLP  (WMMA x2) =====
  mm_tiles(edge_attr, 128, mlp_w1, 64, sHid, 64, 1, 4, 128, wave, lane);
  __syncthreads();
  for (int i = tid; i < 1024; i += 256)
    sHid[i] = fmaxf(sHid[i] + mlp_b1[i & 63], 0.f);
  __syncthreads();
  mm_tiles(sHid, 64, mlp_w2, 64, sEAn, 64, 1, 4, 64, wave, lane);
  __syncthreads();
  for (int i = tid; i < 1024; i += 256)
    sEAn[i] += mlp_b2[i & 63];
  __syncthreads();

  // ===== Stage 6: GAT2 =====
  mm_tiles(sX2, 256, g2_lin, 64, sH2, 64, 1, 4, 256, wave, lane);  // H2 = X2@g2_lin
  if (tid < 64) {            // ea_new column means
    float s = 0.f;
    for (int e = 0; e < 16; ++e) s += sEAn[e * 64 + tid];
    s_mean[tid] = s * (1.f / 16.f);
  }
  if (tid >= 64 && tid < 128) {  // v2[k] = g2_le @ a_e folded
    int k = tid - 64;
    float s = 0.f;
    for (int c = 0; c < 64; ++c) s += g2_le[k * 64 + c] * g2_ae[c];
    s_v[k] = s;
  }
  __syncthreads();
  for (int i = tid; i < 1536; i += 256) {  // ea2_full [24][64]
    int e = i >> 6, c = i & 63;
    sEA2[i] = (e < 16) ? sEAn[i] : s_mean[c];
  }
  if (tid < 8) {
    float sa = 0.f, sd = 0.f;
    for (int c = 0; c < 64; ++c) {
      float h = sH2[tid * 64 + c];
      sa += h * g2_as[c];
      sd += h * g2_ad[c];
    }
    s_as[tid] = sa; s_ad[tid] = sd;
  }
  __syncthreads();
  if (tid < 24) {
    float s = 0.f;
    for (int k = 0; k < 64; ++k) s += sEA2[tid * 64 + k] * s_v[k];
    float a = s_as[s_src[tid]] + s_ad[s_dst[tid]] + s;
    s_ae[tid] = (a > 0.f) ? a : 0.2f * a;
  }
  __syncthreads();
  if (tid < 8) {
    float m = -3.0e38f;
    for (int e = 0; e < 24; ++e) if (s_dst[e] == tid) m = fmaxf(m, s_ae[e]);
    s_mx[tid] = m;
  }
  __syncthreads();
  if (tid < 24) s_w[tid] = __expf(s_ae[tid] - s_mx[s_dst[tid]]);
  __syncthreads();
  if (tid < 8) {
    float s = 0.f;
    for (int e = 0; e < 24; ++e) if (s_dst[e] == tid) s += s_w[e];
    s_den[tid] = s;
  }
  __syncthreads();
  if (tid < 24) s_w[tid] /= (s_den[s_dst[tid]] + 1e-16f);
  __syncthreads();
  for (int i = tid; i < 512; i += 256) {   // aggregate -> X3 (+bias, relu)
    int n = i >> 6, c = i & 63;
    float s = 0.f;
    for (int e = 0; e < 24; ++e)
      if (s_dst[e] == n) s += s_w[e] * sH2[s_src[e] * 64 + c];
    sX3[i] = fmaxf(s + g2_b[c], 0.f);
  }
  __syncthreads();

  // ===== Stage 7: means + deconv einsums -> CNN_2 input [64][4][10] =====
  if (tid < 192) {
    int c = tid & 63;
    float s = 0.f;
    if (tid < 64)       { for (int n = 0; n < 8; ++n) s += sX3[n * 64 + c]; }
    else if (tid < 128) { for (int r = 0; r < 8; ++r) s += sEAn[(2 * r) * 64 + c]; }
    else                { for (int r = 0; r < 8; ++r) s += sEAn[(2 * r + 1) * 64 + c]; }
    s_vec3[tid] = s * 0.125f;
  }
  __syncthreads();
  for (int i = tid; i < 2560; i += 256) {
    int p = i / 40, r = i % 40, ch = r / 10, tt = r % 10;
    float v;
    if (ch == 1) v = xft[p * 10 + tt];
    else {
      const float* dw = (ch == 0) ? d1w : (ch == 2) ? d2w : d3w;
      const float* db = (ch == 0) ? d1b : (ch == 2) ? d2b : d3b;
      const float* mv = s_vec3 + ((ch == 0) ? 0 : (ch == 2) ? 64 : 128);
      float s = db[p];
      for (int j = 0; j < 64; ++j) s += mv[j] * dw[j * 640 + p * 10 + tt];
      v = s;
    }
    sC2[i] = v;
  }
  __syncthreads();

  // ===== Stage 8: conv1 (4->32,k3) + maxpool(2) -> sP [64][32][4] =====
  for (int i = tid; i < 8192; i += 256) {
    int p = i >> 7, r = i & 127, co = r >> 2, q = r & 3;
    float m = -3.0e38f;
    for (int j = 0; j < 2; ++j) {
      int l = 2 * q + j;
      float s = c2b1[co];
      for (int ci = 0; ci < 4; ++ci)
        for (int k = 0; k < 3; ++k)
          s += sC2[p * 40 + ci * 10 + l + k] * c2w1[co * 12 + ci * 3 + k];
      m = fmaxf(m, s);
    }
    sP[p * 128 + co * 4 + q] = m;
  }
  __syncthreads();

  // ===== Stage 9: conv2 (32->64,k3) in-place per row: sP row -> flat [64][128] =====
  for (int p = wave; p < 64; p += 8) {     // rows private to each wave
    float* row = sP + p * 128;
    float vals[4];
#pragma unroll
    for (int jj = 0; jj < 4; ++jj) {       // all loads of this row...
      int col = lane * 4 + jj;
      int co = col >> 1, l2 = col & 1;
      float s = c2b2[co];
      for (int ci = 0; ci < 32; ++ci)
        for (int k = 0; k < 3; ++k)
          s += row[ci * 4 + l2 + k] * c2w2[co * 96 + ci * 3 + k];
      vals[jj] = s;
    }
#pragma unroll
    for (int jj = 0; jj < 4; ++jj)         // ...precede all stores (aliasing order kept)
      row[lane * 4 + jj] = vals[jj];
  }
  __syncthreads();

  // ===== Stage 10: lin1 (WMMA, per 16-row block) + lin2 (WMMA, N padded 10->16) =====
  for (int mt = 0; mt < 4; ++mt) {
    mm_tiles(sP + mt * 2048, 128, c2l1w, 256, sTmp, 256, 1, 16, 128, wave, lane);
    __syncthreads();
    for (int i = tid; i < 4096; i += 256) sTmp[i] += c2l1b[i & 255];  // no relu here
    __syncthreads();
    if (wave == 0) {                      // single 16x16 output tile, cols 0..9 valid
      const int col = lane & 15, g = lane >> 4;
#if USE_WMMA_F32
      v8f acc = {};
      const float* Ar = sTmp + col * 256;
      for (int k = 0; k < 256; k += 4) {
        int ka = k + 2 * g;
        v2f a, b;
        a.x = Ar[ka]; a.y = Ar[ka + 1];
        b.x = (col < 10) ? c2l2w[ka * 10 + col] : 0.f;
        b.y = (col < 10) ? c2l2w[(ka + 1) * 10 + col] : 0.f;
        acc = __builtin_amdgcn_wmma_f32_16x16x4_f32(
            false, a, false, b, (short)0, acc, false, false);
      }
      if (col < 10) {
#pragma unroll
        for (int j = 0; j < 8; ++j) {
          int rowg = mt * 16 + 8 * g + j;
          out[rowg * 10 + col] = fmaxf(acc[j] + c2l2b[col], 0.f);
        }
      }
#else
      for (int j = 0; j < 8; ++j) {
        int m = 8 * g + j;
        if (col < 10) {
          float s = 0.f;
          for (int k = 0; k < 256; ++k) s += sTmp[m * 256 + k] * c2l2w[k * 10 + col];
          out[(mt * 16 + m) * 10 + col] = fmaxf(s + c2l2b[col], 0.f);
        }
      }
#endif
    }
    __syncthreads();
  }
}

extern "C" void kernel_launch(void* const* d_in, const int* in_sizes, int n_in,
                              void* d_out, int out_size, void* d_ws, size_t ws_size,
                              hipStream_t stream) {
  (void)in_sizes; (void)n_in; (void)out_size; (void)d_ws; (void)ws_size;
  const float* p[37];
  for (int i = 0; i < 37; ++i) p[i] = (const float*)d_in[i];
  const int* eidx = (const int*)d_in[37];
  dtgnn_fused<<<dim3(1), dim3(256), 0, stream>>>(
      p[0], p[1], p[2], p[3], p[4], p[5], p[6], p[7], p[8], p[9], p[10], p[11],
      p[12], p[13], p[14], p[15], p[16], p[17], p[18], p[19], p[20], p[21],
      p[22], p[23], p[24], p[25], p[26], p[27], p[28], p[29], p[30], p[31],
      p[32], p[33], p[34], p[35], p[36], eidx, (float*)d_out);
}